// KernelGaussCascadeConv_80668075754011
// MI455X (gfx1250) — compile-verified
//
#include <hip/hip_runtime.h>
#include <math.h>

typedef __attribute__((ext_vector_type(2))) float v2f;
typedef __attribute__((ext_vector_type(8))) float v8f;

#define N_FILT   80
#define FILT_DIM 251
#define KP       252      // K padded to multiple of 4
#define FS_F     16000.0f
#define LX       64000
#define NOUT     63750    // 64000 - 251 + 1
#define NB       8
#define PI_F     3.14159265358979323846f

// ---------------- block reductions (256 threads) ----------------
__device__ __forceinline__ float blk_min(float v, float* red) {
    int t = threadIdx.x;
    red[t] = v; __syncthreads();
    for (int s = 128; s > 0; s >>= 1) {
        if (t < s) red[t] = fminf(red[t], red[t + s]);
        __syncthreads();
    }
    float r = red[0]; __syncthreads(); return r;
}
__device__ __forceinline__ float blk_max(float v, float* red) {
    int t = threadIdx.x;
    red[t] = v; __syncthreads();
    for (int s = 128; s > 0; s >>= 1) {
        if (t < s) red[t] = fmaxf(red[t], red[t + s]);
        __syncthreads();
    }
    float r = red[0]; __syncthreads(); return r;
}
__device__ __forceinline__ float blk_sum(float v, float* red) {
    int t = threadIdx.x;
    red[t] = v; __syncthreads();
    for (int s = 128; s > 0; s >>= 1) {
        if (t < s) red[t] = red[t] + red[t + s];
        __syncthreads();
    }
    float r = red[0]; __syncthreads(); return r;
}

// ir = 2*(ir-mn)/(mx-mn+1e-6) - 1 ; ir -= mean(ir)   (over 251 valid lanes)
__device__ __forceinline__ float norm_pm1_val(float v, bool valid, float* red) {
    float mn = blk_min(valid ? v :  3.402823466e38f, red);
    float mx = blk_max(valid ? v : -3.402823466e38f, red);
    v = 2.0f * (v - mn) / (mx - mn + 1e-6f) - 1.0f;
    float mean = blk_sum(valid ? v : 0.0f, red) * (1.0f / (float)FILT_DIM);
    return v - mean;
}

// ---------------- filter-bank construction (tiny) ----------------
__global__ __launch_bounds__(256)
void build_filters_kernel(const float* __restrict__ nf1, const float* __restrict__ nf2,
                          const float* __restrict__ nf3, const float* __restrict__ nf4,
                          const float* __restrict__ amp1, const float* __restrict__ amp2,
                          float* __restrict__ Fg) {
    __shared__ float s_ir1[FILT_DIM];
    __shared__ float s_ir2[FILT_DIM];
    __shared__ float red[256];

    const int f = blockIdx.x;
    const int i = threadIdx.x;
    const bool valid = (i < FILT_DIM);

    const float mf = 50.0f / 16000.0f;
    float f1 = fminf(fmaxf(fabsf(nf1[f]) + mf, 0.0f), 0.5f);
    float f2 = fminf(fmaxf(f1 + fabsf(nf2[f] - f1) + mf, 0.0f), 0.5f);
    float f3 = fminf(fmaxf(fabsf(nf3[f]) + mf, 0.0f), 0.5f);
    float f4 = fminf(fmaxf(f3 + fabsf(nf4[f] - f3) + mf, 0.0f), 0.5f);
    float a1 = fabsf(amp1[f]);
    float a2 = fabsf(amp2[f]);

    float t = (float)(i + 1) / FS_F;
    float bw1 = (f2 - f1) * FS_F, fc1 = 0.5f * (f1 + f2) * FS_F;
    float bw2 = (f4 - f3) * FS_F, fc2 = 0.5f * (f3 + f4) * FS_F;
    float g1 = PI_F * bw1, g2 = PI_F * bw2;

    float ir1 = a1 * expf(-2.0f * g1 * g1 * t * t) * cosf(2.0f * PI_F * fc1 * t);
    float ir2 = a2 * expf(-2.0f * g2 * g2 * t * t) * cosf(2.0f * PI_F * fc2 * t);

    ir1 = norm_pm1_val(ir1, valid, red);
    ir2 = norm_pm1_val(ir2, valid, red);

    if (valid) { s_ir1[i] = ir1; s_ir2[i] = ir2; }
    __syncthreads();

    // casc[i] = sum_m ir1_padded[i+m] * ir2[m], pad = 125 zeros each side
    float c = 0.0f;
    if (valid) {
        int jlo = (i - 125 > 0) ? (i - 125) : 0;
        int jhi = (i + 125 < 250) ? (i + 125) : 250;
        for (int j = jlo; j <= jhi; ++j)
            c += s_ir1[j] * s_ir2[j - i + 125];
    }
    c = norm_pm1_val(c, valid, red);

    if (valid) {
        // window[i] = 0.54 - 0.46*cos(2*pi * (linspace(0,251,251)[i]) / 251)
        //           = 0.54 - 0.46*cos(2*pi * i / 250)
        float w = 0.54f - 0.46f * cosf(2.0f * PI_F * (float)i / 250.0f);
        Fg[f * KP + i] = c * w;
    }
    if (i == FILT_DIM) Fg[f * KP + FILT_DIM] = 0.0f;  // K pad
}

// ---------------- implicit-GEMM conv via V_WMMA_F32_16X16X4_F32 ----------------
// Block: 256 threads = 8 waves. Block tile: M=16 filters, N=512 outputs.
// Wave tile: 16x64 (4 accumulators). K chunked by 64 through LDS.
__global__ __launch_bounds__(256)
void conv_wmma_kernel(const float* __restrict__ x, const float* __restrict__ Fg,
                      float* __restrict__ out) {
    __shared__ float sX[768];          // x window: 512 outputs + K reach
    __shared__ float sF[64 * 16];      // filter chunk, [k][m] layout

    const int tid  = threadIdx.x;
    const int lane = tid & 31;
    const int wave = tid >> 5;
    const int b    = blockIdx.z;
    const int m0   = blockIdx.y * 16;
    const int n0   = blockIdx.x * 512;

    const int l15 = lane & 15;
    const int kh  = lane >> 4;        // 0: K pair k4,k4+1; 1: K pair k4+2,k4+3 (per WMMA layout)
    const int wbase = wave * 64;

    // stage x window (zero-fill past end of signal)
    const float* xb = x + (size_t)b * LX;
    for (int i = tid; i < 768; i += 256) {
        int gi = n0 + i;
        sX[i] = (gi < LX) ? xb[gi] : 0.0f;
    }

    v8f acc0 = {}, acc1 = {}, acc2 = {}, acc3 = {};

    for (int kc = 0; kc < KP; kc += 64) {
        const int klen = (KP - kc < 64) ? (KP - kc) : 64;  // 64,64,64,60
        __syncthreads();
        // stage filter chunk into [k][m] layout (conflict-free A reads)
        for (int idx = tid; idx < klen * 16; idx += 256) {
            int k = idx >> 4, m = idx & 15;
            int gk = kc + k;
            sF[idx] = (gk < FILT_DIM) ? Fg[(size_t)(m0 + m) * KP + gk] : 0.0f;
        }
        __syncthreads();

        for (int k4 = 0; k4 < klen; k4 += 4) {
            // A fragment (16x4 f32): lane -> (M = lane&15, K = k4 + 2*kh {,+1})
            v2f a;
            a[0] = sF[(k4 + 2 * kh)     * 16 + l15];
            a[1] = sF[(k4 + 2 * kh + 1) * 16 + l15];
            const int gk = kc + k4;
            // B fragments (4x16 f32): lane -> (K = 2*kh {,+1}, N = lane&15)
            // B[k][n] = x[n0 + n + k]  (im2col row = shifted signal)
            const int base = wbase + l15 + 2 * kh + gk;
            v2f b0; b0[0] = sX[base];        b0[1] = sX[base + 1];
            v2f b1; b1[0] = sX[base + 16];   b1[1] = sX[base + 17];
            v2f b2; b2[0] = sX[base + 32];   b2[1] = sX[base + 33];
            v2f b3; b3[0] = sX[base + 48];   b3[1] = sX[base + 49];

            acc0 = __builtin_amdgcn_wmma_f32_16x16x4_f32(false, a, false, b0, (short)0, acc0, false, false);
            acc1 = __builtin_amdgcn_wmma_f32_16x16x4_f32(false, a, false, b1, (short)0, acc1, false, false);
            acc2 = __builtin_amdgcn_wmma_f32_16x16x4_f32(false, a, false, b2, (short)0, acc2, false, false);
            acc3 = __builtin_amdgcn_wmma_f32_16x16x4_f32(false, a, false, b3, (short)0, acc3, false, false);
        }
    }

    // store: C/D layout -> VGPR r: (M = r + 8*kh, N = lane&15)
    float* ob = out + (size_t)(b * N_FILT + m0) * NOUT;
    v8f accs[4] = {acc0, acc1, acc2, acc3};
    #pragma unroll
    for (int ns = 0; ns < 4; ++ns) {
        int n = n0 + wbase + ns * 16 + l15;
        if (n < NOUT) {
            #pragma unroll
            for (int r = 0; r < 8; ++r) {
                int m = r + 8 * kh;
                ob[(size_t)m * NOUT + n] = accs[ns][r];
            }
        }
    }
}

extern "C" void kernel_launch(void* const* d_in, const int* in_sizes, int n_in,
                              void* d_out, int out_size, void* d_ws, size_t ws_size,
                              hipStream_t stream) {
    const float* x    = (const float*)d_in[0];
    const float* nf1  = (const float*)d_in[1];
    const float* nf2  = (const float*)d_in[2];
    const float* nf3  = (const float*)d_in[3];
    const float* nf4  = (const float*)d_in[4];
    const float* amp1 = (const float*)d_in[5];
    const float* amp2 = (const float*)d_in[6];
    float* Fg = (float*)d_ws;                 // 80 * 252 floats
    float* out = (float*)d_out;

    build_filters_kernel<<<N_FILT, 256, 0, stream>>>(nf1, nf2, nf3, nf4, amp1, amp2, Fg);

    dim3 grid((NOUT + 511) / 512, N_FILT / 16, NB);   // 125 x 5 x 8
    conv_wmma_kernel<<<grid, 256, 0, stream>>>(x, Fg, out);
}